// IoU_65429531787858
// MI455X (gfx1250) — compile-verified
//
#include <hip/hip_runtime.h>
#include <stdint.h>

// Problem constants (match reference setup_inputs)
#define CCLS 21
#define BB   16
#define HWPX (512 * 512)            // 262144 pixels per (b,c) plane
#define TPB  128
#define PXT  4                      // pixels per thread per iteration (b128)
#define TILE (TPB * PXT)            // 512 pixels per block-iteration
#define CHUNKS_PER_IMG 128
#define PX_PER_CHUNK (HWPX / CHUNKS_PER_IMG)   // 2048
#define ITERS (PX_PER_CHUNK / TILE)            // 4
#define EPSV 1e-7f

// Workspace layout (floats): [0,336) inter  [336,672) gt  [672,1008) pr
#define NCELL (BB * CCLS)

typedef float v4f __attribute__((ext_vector_type(4)));

__global__ void iou_zero_ws(float* __restrict__ ws) {
    int i = blockIdx.x * blockDim.x + threadIdx.x;
    if (i < 3 * NCELL) ws[i] = 0.0f;
}

__global__ __launch_bounds__(TPB) void iou_phase1(const float* __restrict__ inp,
                                                  const float* __restrict__ tgt,
                                                  float* __restrict__ ws) {
    // Double-buffered target tile streamed by CDNA5 async-to-LDS DMA (B128).
    // 2 * 21 * 512 * 4B = 84KB (gfx1250 allows up to 320KB LDS per workgroup)
    __shared__ float tile[2][CCLS][TILE];
    __shared__ float s_inter[CCLS];
    __shared__ float s_pr[CCLS];
    __shared__ float s_gt[CCLS];

    const int t     = threadIdx.x;
    const int b     = blockIdx.x >> 7;     // 128 chunks per image
    const int chunk = blockIdx.x & 127;

    if (t < CCLS) { s_inter[t] = 0.0f; s_pr[t] = 0.0f; s_gt[t] = 0.0f; }
    __syncthreads();

    const size_t img_off = (size_t)b * CCLS * HWPX;
    const float* ibase = inp + img_off;
    const float* tbase = tgt + img_off;
    const int    px0   = chunk * PX_PER_CHUNK;

    float gt_acc[CCLS];
#pragma unroll
    for (int c = 0; c < CCLS; ++c) gt_acc[c] = 0.0f;

    // Issue 21 async B128 loads (one per class plane) for iteration `it` into
    // buffer `buf`.  Per-wave ASYNCcnt tracks these; each lane supplies its own
    // LDS byte offset (VDST) and 64-bit global address (VADDR).
    // gfx1250: GLOBAL_LOAD_ASYNC_TO_LDS_B128 (512B per wave per instruction).
    auto issue = [&](int it, int buf) {
        const int px = px0 + it * TILE + t * PXT;
        const float* g = tbase + px;
        uint32_t lds = (uint32_t)(uintptr_t)(&tile[buf][0][t * PXT]);
#pragma unroll
        for (int c = 0; c < CCLS; ++c) {
            asm volatile("global_load_async_to_lds_b128 %0, %1, off"
                         :: "v"(lds), "v"(g)
                         : "memory");
            g += HWPX;
            lds += TILE * 4;
        }
    };

    issue(0, 0);

    for (int it = 0; it < ITERS; ++it) {
        const int buf = it & 1;
        const int px  = px0 + it * TILE + t * PXT;

        // ---- argmax over 21 logit planes for 4 pixels, b128 nontemporal streams
        const v4f* ip = (const v4f*)(ibase + px);
        v4f  best = __builtin_nontemporal_load(ip);
        int  bci[PXT] = {0, 0, 0, 0};
#pragma unroll
        for (int c = 1; c < CCLS; ++c) {
            v4f v = __builtin_nontemporal_load(ip + (size_t)c * (HWPX / PXT));
#pragma unroll
            for (int j = 0; j < PXT; ++j) {
                if (v[j] > best[j]) { best[j] = v[j]; bci[j] = c; }  // first-max argmax
            }
        }

        // ---- prefetch next tile, then wait for the current one (in-order per-wave
        //      ASYNCcnt; each lane reads only LDS words its own wave requested)
        if (it + 1 < ITERS) {
            issue(it + 1, (it + 1) & 1);
            asm volatile("s_wait_asynccnt 21" ::: "memory");
        } else {
            asm volatile("s_wait_asynccnt 0" ::: "memory");
        }

        // ---- accumulate: gt sums in registers, select target value at argmax class
        float sel[PXT] = {0.0f, 0.0f, 0.0f, 0.0f};
#pragma unroll
        for (int c = 0; c < CCLS; ++c) {
            v4f tv = *(const v4f*)&tile[buf][c][t * PXT];   // ds_load_b128
            gt_acc[c] += (tv[0] + tv[1]) + (tv[2] + tv[3]);
#pragma unroll
            for (int j = 0; j < PXT; ++j) sel[j] = (c == bci[j]) ? tv[j] : sel[j];
        }
#pragma unroll
        for (int j = 0; j < PXT; ++j) {
            atomicAdd(&s_inter[bci[j]], sel[j]);   // ds_add_f32 (exact: integer-valued)
            atomicAdd(&s_pr[bci[j]], 1.0f);
        }
    }

    // fold per-thread gt accumulators into LDS
#pragma unroll
    for (int c = 0; c < CCLS; ++c) atomicAdd(&s_gt[c], gt_acc[c]);
    __syncthreads();

    if (t < CCLS) {
        const int cell = b * CCLS + t;
        atomicAdd(&ws[cell],             s_inter[t]);
        atomicAdd(&ws[NCELL + cell],     s_gt[t]);
        atomicAdd(&ws[2 * NCELL + cell], s_pr[t]);
    }
}

__global__ void iou_phase2(const float* __restrict__ ws, float* __restrict__ out) {
    const int t = threadIdx.x;
    if (t >= CCLS - 1) return;
    const int c = t + 1;  // skip background class 0
    float inter_s = 0.0f, union_acc = 0.0f, cnt = 0.0f;
    for (int b = 0; b < BB; ++b) {
        float inter = ws[b * CCLS + c];
        float gt    = ws[NCELL + b * CCLS + c];
        float pr    = ws[2 * NCELL + b * CCLS + c];
        // target is binary -> any(target != 0) <=> exact integer gt_sum > 0
        float present = (gt != 0.0f) ? 1.0f : 0.0f;
        inter_s   += present * inter;
        union_acc += present * (gt + pr);
        cnt       += present;
    }
    float denom = union_acc - inter_s + EPSV;
    out[t]      = (inter_s / denom) * cnt;  // scores[1:]
    out[20 + t] = cnt;                      // counts[1:]
}

extern "C" void kernel_launch(void* const* d_in, const int* in_sizes, int n_in,
                              void* d_out, int out_size, void* d_ws, size_t ws_size,
                              hipStream_t stream) {
    const float* inp = (const float*)d_in[0];
    const float* tgt = (const float*)d_in[1];
    float* ws  = (float*)d_ws;
    float* out = (float*)d_out;

    iou_zero_ws<<<1, 1024, 0, stream>>>(ws);
    iou_phase1<<<BB * CHUNKS_PER_IMG, TPB, 0, stream>>>(inp, tgt, ws);
    iou_phase2<<<1, 32, 0, stream>>>(ws, out);
}